// DiceLoss_30245159698510
// MI455X (gfx1250) — compile-verified
//
#include <hip/hip_runtime.h>
#include <hip/hip_bf16.h>
#include <math.h>

#define IMG_H 1024
#define IMG_W 1024
#define KR    31
#define PADR  15            // KR/2
#define TW    128           // tile width  (outputs)
#define TH    64            // tile height (outputs)
#define HHALO (TH + KR - 1) // 94 halo rows
#define WHALO (TW + KR - 1) // 158 halo cols
#define HSTR  (WHALO + 1)   // 159, odd LDS row stride -> conflict-free column walks
#define NTHREADS 256
#define SMOOTH_F 0.001f

#define AS1 __attribute__((address_space(1)))
#define AS3 __attribute__((address_space(3)))

// ---- CDNA5 async global->LDS path (guarded so compile stays green) ----
#ifndef USE_ASYNC_LDS
#if defined(__has_builtin)
#if __has_builtin(__builtin_amdgcn_global_load_async_to_lds_b32) && \
    __has_builtin(__builtin_amdgcn_s_wait_asynccnt)
#define USE_ASYNC_LDS 1
#endif
#endif
#endif
#ifndef USE_ASYNC_LDS
#define USE_ASYNC_LDS 0
#endif

__global__ __launch_bounds__(NTHREADS)
void dice_tile_kernel(const float* __restrict__ logits,
                      const float* __restrict__ target,
                      float* __restrict__ acc)
{
    __shared__ float halo[HHALO * HSTR];   // ~58.4 KB
    __shared__ float red[2 * (NTHREADS / 32)];

    const int tid    = threadIdx.x;
    const int tile_x = blockIdx.x * TW;
    const int tile_y = blockIdx.y * TH;
    const int img    = blockIdx.z;

    const float* __restrict__ tgt = target + (size_t)img * (IMG_H * IMG_W);
    const float* __restrict__ lgt = logits + (size_t)img * (IMG_H * IMG_W);

    // ---------- Phase 0: fill halo tile of target into LDS (zero-padded) ----------
    for (int i = tid; i < HHALO * WHALO; i += NTHREADS) {
        int r = i / WHALO;
        int c = i - r * WHALO;
        int gy = tile_y - PADR + r;
        int gx = tile_x - PADR + c;
        bool inb = ((unsigned)gy < (unsigned)IMG_H) & ((unsigned)gx < (unsigned)IMG_W);
        int lofs = r * HSTR + c;
#if USE_ASYNC_LDS
        if (inb) {
            __builtin_amdgcn_global_load_async_to_lds_b32(
                (AS1 int*)(uintptr_t)(tgt + (size_t)gy * IMG_W + gx),
                (AS3 int*)(AS3 float*)(&halo[lofs]),
                0, 0);
        } else {
            halo[lofs] = 0.0f;
        }
#else
        halo[lofs] = inb ? tgt[(size_t)gy * IMG_W + gx] : 0.0f;
#endif
    }

    // Warm the logits tile (global_prefetch_b8) while the halo fill is in flight.
    {
        int py = tile_y + (tid >> 2);
        int px = tile_x + (tid & 3) * 32;
        __builtin_prefetch(&lgt[(size_t)py * IMG_W + px], 0, 0);
    }

#if USE_ASYNC_LDS
    __builtin_amdgcn_s_wait_asynccnt(0);
#endif
    __syncthreads();

    // ---------- Phase 1: vertical 31-tap box sums, in place (one thread/column) ----------
    if (tid < WHALO) {
        const int c = tid;
        float s = 0.0f;
#pragma unroll
        for (int r = 0; r < KR; ++r) s += halo[r * HSTR + c];
        for (int y = 0;; ++y) {
            float outgoing = halo[y * HSTR + c];   // original value of row y
            halo[y * HSTR + c] = s;                // vsum[y] overwrites row y (safe)
            if (y + 1 >= TH) break;
            s += halo[(y + KR) * HSTR + c] - outgoing;
        }
    }
    __syncthreads();

    // ---------- Phase 2: horizontal 31-tap sliding sum fused with loss terms ----------
    const int   y    = tid >> 2;          // 0..63
    const int   xs   = (tid & 3) * 32;    // 0,32,64,96
    const int   gy   = tile_y + y;
    const float* __restrict__ hrow = &halo[y * HSTR];

    float s = 0.0f;
#pragma unroll
    for (int j = 0; j < KR; ++j) s += hrow[xs + j];

    float inter = 0.0f, msum = 0.0f;
    const float inv = 1.0f / (float)(KR * KR);
    const size_t rowbase = (size_t)gy * IMG_W + tile_x;
#pragma unroll 4
    for (int i = 0; i < 32; ++i) {
        int   x      = xs + i;
        float pooled = s * inv;
        float t      = tgt[rowbase + x];
        float z      = lgt[rowbase + x];
        float o      = 1.0f / (1.0f + __expf(-z));
        float w      = fmaf(5.0f, fabsf(pooled - t), 1.0f);
        inter = fmaf(o * t, w, inter);
        msum  = fmaf(t + o, w, msum);
        if (i + 1 < 32) s += hrow[x + KR] - hrow[x];
    }

    // ---------- Phase 3: wave32 shuffle tree + cross-wave LDS + one atomic pair ----------
#pragma unroll
    for (int off = 16; off > 0; off >>= 1) {
        inter += __shfl_down(inter, off);
        msum  += __shfl_down(msum,  off);
    }
    const int wid = tid >> 5, lane = tid & 31;
    if (lane == 0) { red[wid * 2] = inter; red[wid * 2 + 1] = msum; }
    __syncthreads();
    if (tid == 0) {
        float I = 0.0f, M = 0.0f;
#pragma unroll
        for (int w8 = 0; w8 < NTHREADS / 32; ++w8) { I += red[2 * w8]; M += red[2 * w8 + 1]; }
        atomicAdd(&acc[img * 2 + 0], I);
        atomicAdd(&acc[img * 2 + 1], M);
    }
}

__global__ void dice_finalize_kernel(const float* __restrict__ acc,
                                     float* __restrict__ out, int B)
{
    int t = threadIdx.x;
    float l = 0.0f;
    for (int b = t; b < B; b += 32) {
        float I = acc[2 * b], M = acc[2 * b + 1];
        l += 1.0f - (2.0f * I + SMOOTH_F) / (M + SMOOTH_F);
    }
#pragma unroll
    for (int off = 16; off > 0; off >>= 1) l += __shfl_down(l, off);
    if (t == 0) out[0] = l / (float)B;
}

extern "C" void kernel_launch(void* const* d_in, const int* in_sizes, int n_in,
                              void* d_out, int out_size, void* d_ws, size_t ws_size,
                              hipStream_t stream)
{
    (void)n_in; (void)out_size; (void)ws_size;
    const float* logits = (const float*)d_in[0];  // "output" (pre-sigmoid)
    const float* target = (const float*)d_in[1];  // "target"
    float* acc = (float*)d_ws;                    // 2 floats per image
    float* out = (float*)d_out;                   // scalar loss

    const int B = in_sizes[0] / (IMG_H * IMG_W);  // 32

    (void)hipMemsetAsync(d_ws, 0, (size_t)(2 * B) * sizeof(float), stream);

    dim3 grid(IMG_W / TW, IMG_H / TH, B);         // 8 x 16 x 32 = 4096 tiles
    dice_tile_kernel<<<grid, NTHREADS, 0, stream>>>(logits, target, acc);
    dice_finalize_kernel<<<1, 32, 0, stream>>>(acc, out, B);
}